// MSSSIM_39582418600385
// MI455X (gfx1250) — compile-verified
//
#include <hip/hip_runtime.h>
#include <math.h>

typedef __attribute__((ext_vector_type(2))) float v2f;
typedef __attribute__((ext_vector_type(8))) float v8f;

// 11-tap Gaussian, sigma=1.5, normalized (matches reference _gaussian_window)
__device__ __constant__ float G11[11] = {
    0.00102838f, 0.00759876f, 0.03600077f, 0.10936069f, 0.21300553f,
    0.26601172f,
    0.21300553f, 0.10936069f, 0.03600077f, 0.00759876f, 0.00102838f};

static __device__ inline v2f mk2(float a, float b) { v2f v; v[0] = a; v[1] = b; return v; }

#define WMMA4(A, B, C) \
    __builtin_amdgcn_wmma_f32_16x16x4_f32(false, (A), false, (B), (short)0, (C), false, false)

#define C1_SSIM 1.0e-4f
#define C2_SSIM 9.0e-4f

// One MS-SSIM pyramid level: fused separable 11x11 Gaussian blur of
// (x1, x2, x1^2, x2^2, x1*x2) via f32 WMMA, per-pixel ssim/mcs maps,
// block-reduced sums atomically accumulated into ssim_acc / mcs_acc.
// Block: 256 threads (8 waves); tile: 32 rows x 64 cols of output per block.
__global__ __launch_bounds__(256) void ssim_level_kernel(
    const float* __restrict__ img1, const float* __restrict__ img2,
    int H, int W, float* __restrict__ ssim_acc, float* __restrict__ mcs_acc) {
  constexpr int IP = 76;  // input LDS pitch (64 + 10 halo + pad; 76%64=12 -> conflict-free strided reads)
  constexpr int IR = 48;  // input LDS rows (32 out + 10 halo + pad to 3x16)
  constexpr int PM = 72;  // mid LDS pitch (72%64=8 -> conflict-free K-pair reads)

  __shared__ float s_in1[IR * IP];
  __shared__ float s_in2[IR * IP];
  __shared__ float s_midA[IR * PM];
  __shared__ float s_midB[IR * PM];
  __shared__ float s_red[2];

  const int tid = threadIdx.x;
  const int lane = tid & 31;
  // wave id forced into an SGPR: all tile-selection control flow becomes
  // scalar branches, guaranteeing EXEC == all-ones around every WMMA.
  const int wvu = __builtin_amdgcn_readfirstlane(tid >> 5);  // 0..7
  const int lo = lane & 15;   // M (A) / N (B,C/D)
  const int hi = lane >> 4;   // K-half selector

  const int col0 = blockIdx.x * 64;
  const int row0 = blockIdx.y * 32;
  const size_t plane = (size_t)H * (size_t)W;
  const float* p1 = img1 + (size_t)blockIdx.z * plane;
  const float* p2 = img2 + (size_t)blockIdx.z * plane;

  if (tid < 2) s_red[tid] = 0.0f;

  // Stage inputs + halo into LDS, zero padded (== conv SAME).
  for (int idx = tid; idx < IR * IP; idx += 256) {
    int h = idx / IP, w = idx - h * IP;
    int gy = row0 - 5 + h, gx = col0 - 5 + w;
    bool ok = (gy >= 0) & (gy < H) & (gx >= 0) & (gx < W);
    size_t o = (size_t)gy * W + gx;
    s_in1[idx] = ok ? p1[o] : 0.0f;
    s_in2[idx] = ok ? p2[o] : 0.0f;
  }

  // Per-lane banded Gaussian chunks. Same values serve as B (horizontal pass,
  // value g[K-N]) and as A (vertical pass, value g[K-M]) since M==N==lane&15.
  float wB[7][2];
#pragma unroll
  for (int c4 = 0; c4 < 7; ++c4) {
#pragma unroll
    for (int kk = 0; kk < 2; ++kk) {
      int K = 4 * c4 + 2 * hi + kk;
      int d = K - lo;
      wB[c4][kk] = (d >= 0 && d <= 10) ? G11[d] : 0.0f;
    }
  }

  __syncthreads();

  const int R2 = wvu >> 2;  // output 16-row tile (0..1)
  const int C2 = wvu & 3;   // output 16-col tile (0..3)
  const int vcol = 16 * C2 + lo;           // vertical-pass mid column
  const int vkbase = 16 * R2 + 2 * hi;     // vertical-pass mid row base

  v8f vmu1 = {}, vmu2 = {}, vb11 = {}, vb22 = {}, vb12 = {};

#pragma unroll
  for (int phase = 0; phase < 3; ++phase) {
    // ---- horizontal pass: 12 mid tiles (3 row-tiles x 4 col-tiles) ----
    // wave w handles tile w, and (waves 0..3) tile w+8: uniform scalar branch.
    auto horiz_tile = [&](int t) {
      int r = t >> 2, c = t & 3;
      v8f acc0 = {}, acc1 = {};
      int rowbase = (16 * r + lo) * IP + 16 * c + 2 * hi;
#pragma unroll
      for (int c4 = 0; c4 < 7; ++c4) {
        int cb = rowbase + 4 * c4;
        float x1a = s_in1[cb], x1b = s_in1[cb + 1];
        float x2a = s_in2[cb], x2b = s_in2[cb + 1];
        v2f B = mk2(wB[c4][0], wB[c4][1]);
        v2f A0, A1;
        if (phase == 0) {
          A0 = mk2(x1a, x1b);
          A1 = mk2(x2a, x2b);
        } else if (phase == 1) {
          A0 = mk2(x1a * x1a, x1b * x1b);
          A1 = mk2(x2a * x2a, x2b * x2b);
        } else {
          A0 = mk2(x1a * x2a, x1b * x2b);
          A1 = A0;
        }
        acc0 = WMMA4(A0, B, acc0);
        if (phase != 2) acc1 = WMMA4(A1, B, acc1);
      }
      int obase = (16 * r + 8 * hi) * PM + 16 * c + lo;
#pragma unroll
      for (int v = 0; v < 8; ++v) {
        int o = obase + v * PM;
        s_midA[o] = acc0[v];
        if (phase != 2) s_midB[o] = acc1[v];
      }
    };
    horiz_tile(wvu);
    if (wvu < 4) horiz_tile(wvu + 8);
    __syncthreads();

    // ---- vertical pass: one 16x16 output tile per wave ----
    {
      v8f o0 = {}, o1 = {};
#pragma unroll
      for (int c4 = 0; c4 < 7; ++c4) {
        int kb = vkbase + 4 * c4;
        int mo = kb * PM + vcol;
        v2f Av = mk2(wB[c4][0], wB[c4][1]);
        v2f B0 = mk2(s_midA[mo], s_midA[mo + PM]);
        o0 = WMMA4(Av, B0, o0);
        if (phase != 2) {
          v2f B1 = mk2(s_midB[mo], s_midB[mo + PM]);
          o1 = WMMA4(Av, B1, o1);
        }
      }
      if (phase == 0) { vmu1 = o0; vmu2 = o1; }
      else if (phase == 1) { vb11 = o0; vb22 = o1; }
      else { vb12 = o0; }
    }
    __syncthreads();  // before mid buffers are reused
  }

  // ---- elementwise ssim/mcs + reduction ----
  float ss = 0.0f, ms = 0.0f;
  const int gx = col0 + vcol;
  const int gybase = row0 + 16 * R2 + 8 * hi;
#pragma unroll
  for (int v = 0; v < 8; ++v) {
    float mu1 = vmu1[v], mu2 = vmu2[v];
    float mu1sq = mu1 * mu1, mu2sq = mu2 * mu2, mu12 = mu1 * mu2;
    float s1 = vb11[v] - mu1sq;
    float s2 = vb22[v] - mu2sq;
    float s12 = vb12[v] - mu12;
    float mcs = (2.0f * s12 + C2_SSIM) / (s1 + s2 + C2_SSIM);
    float ssim = ((2.0f * mu12 + C1_SSIM) / (mu1sq + mu2sq + C1_SSIM)) * mcs;
    if ((gybase + v) < H && gx < W) { ss += ssim; ms += mcs; }
  }
#pragma unroll
  for (int off = 16; off > 0; off >>= 1) {
    ss += __shfl_xor(ss, off, 32);
    ms += __shfl_xor(ms, off, 32);
  }
  if (lane == 0) { atomicAdd(&s_red[0], ss); atomicAdd(&s_red[1], ms); }
  __syncthreads();
  if (tid == 0) { atomicAdd(ssim_acc, s_red[0]); atomicAdd(mcs_acc, s_red[1]); }
}

// 2x2 average pool of both images (NCHW, C=1, N=32).
__global__ __launch_bounds__(256) void pool2_kernel(
    const float* __restrict__ i1, const float* __restrict__ i2,
    float* __restrict__ o1, float* __restrict__ o2, int OH, int OW) {
  int idx = blockIdx.x * 256 + threadIdx.x;
  int total = 32 * OH * OW;
  if (idx >= total) return;
  int x = idx % OW;
  int t = idx / OW;
  int y = t % OH;
  int b = t / OH;
  int IW = 2 * OW;
  size_t base = ((size_t)b * (2 * OH) + 2 * y) * (size_t)IW + 2 * x;
  o1[idx] = 0.25f * (i1[base] + i1[base + 1] + i1[base + IW] + i1[base + IW + 1]);
  o2[idx] = 0.25f * (i2[base] + i2[base + 1] + i2[base + IW] + i2[base + IW + 1]);
}

__global__ void init_acc_kernel(float* acc) {
  if (threadIdx.x < 16) acc[threadIdx.x] = 0.0f;
}

__global__ void finalize_kernel(const float* __restrict__ acc, float* __restrict__ out) {
  if (threadIdx.x == 0 && blockIdx.x == 0) {
    const float wts[5] = {0.0448f, 0.2856f, 0.3001f, 0.2363f, 0.1333f};
    float r = 1.0f;
    int H = 512;
    for (int l = 0; l < 5; ++l) {
      float n = 32.0f * (float)H * (float)H;
      float mcs_m = acc[5 + l] / n;
      float ssim_m = acc[l] / n;
      r *= powf((l < 4) ? mcs_m : ssim_m, wts[l]);
      H >>= 1;
    }
    out[0] = r;
  }
}

extern "C" void kernel_launch(void* const* d_in, const int* in_sizes, int n_in,
                              void* d_out, int out_size, void* d_ws, size_t ws_size,
                              hipStream_t stream) {
  const float* img1 = (const float*)d_in[0];
  const float* img2 = (const float*)d_in[1];
  float* ws = (float*)d_ws;

  // workspace layout (floats): [0..15] accumulators (ssim[5], mcs[5], pad),
  // then pyramid levels 1..4 for both images.
  const size_t S1 = 32ull * 256 * 256;
  const size_t S2 = 32ull * 128 * 128;
  const size_t S3 = 32ull * 64 * 64;
  const size_t S4 = 32ull * 32 * 32;
  float* acc = ws;
  float* a1 = ws + 16;  float* b1 = a1 + S1;
  float* a2 = b1 + S1;  float* b2 = a2 + S2;
  float* a3 = b2 + S2;  float* b3 = a3 + S3;
  float* a4 = b3 + S3;  float* b4 = a4 + S4;

  init_acc_kernel<<<1, 32, 0, stream>>>(acc);

  auto level = [&](const float* x, const float* y, int H, int W, int l) {
    dim3 g((W + 63) / 64, (H + 31) / 32, 32);
    ssim_level_kernel<<<g, 256, 0, stream>>>(x, y, H, W, acc + l, acc + 5 + l);
  };
  auto pool = [&](const float* x, const float* y, float* ox, float* oy, int OH, int OW) {
    int total = 32 * OH * OW;
    pool2_kernel<<<(total + 255) / 256, 256, 0, stream>>>(x, y, ox, oy, OH, OW);
  };

  level(img1, img2, 512, 512, 0);
  pool(img1, img2, a1, b1, 256, 256);
  level(a1, b1, 256, 256, 1);
  pool(a1, b1, a2, b2, 128, 128);
  level(a2, b2, 128, 128, 2);
  pool(a2, b2, a3, b3, 64, 64);
  level(a3, b3, 64, 64, 3);
  pool(a3, b3, a4, b4, 32, 32);
  level(a4, b4, 32, 32, 4);

  finalize_kernel<<<1, 32, 0, stream>>>(acc, (float*)d_out);
}